// DualAttentionEdgeGAT_33663953666664
// MI455X (gfx1250) — compile-verified
//
#include <hip/hip_runtime.h>
#include <hip/hip_bf16.h>
#include <math.h>

typedef float v2f __attribute__((ext_vector_type(2)));
typedef float v8f __attribute__((ext_vector_type(8)));

#define TE 32              // edges (or nodes) per workgroup
#define SA_STRIDE 772      // 768 + 4  (stride % 64 == 4 -> conflict free, %4==0 -> 16B rows)
#define SH_STRIDE 516      // 512 + 4
#define SP_STRIDE 260      // 256 + 4

__device__ __forceinline__ v8f wmma_f32(v8f acc, v2f a, v2f b) {
  // D = A(16x4) * B(4x16) + C, fp32
  return __builtin_amdgcn_wmma_f32_16x16x4_f32(false, a, false, b, (short)0, acc, false, false);
}

// Monotone float max via int atomics (deterministic: mem -> max(mem, val))
__device__ __forceinline__ void atomicMaxFloat(float* addr, float val) {
  if (val >= 0.f) atomicMax((int*)addr, __float_as_int(val));
  else            atomicMin((unsigned int*)addr, (unsigned int)__float_as_int(val));
}

// ---------------------------------------------------------------------------
// CDNA5 async Global->LDS DMA (16B per lane, ASYNCcnt-tracked).
// GV form: per-lane 64-bit global address.  GVS form: SGPR base + 32-bit voff.
// Low 32 bits of a generic LDS pointer are the DS byte address (aperture rule).
// ---------------------------------------------------------------------------
__device__ __forceinline__ void async_ld_b128_gv(const void* gaddr, void* lds) {
  asm volatile("global_load_async_to_lds_b128 %0, %1, off"
               :: "v"((unsigned)(size_t)lds), "v"(gaddr) : "memory");
}
__device__ __forceinline__ void async_ld_b128_gvs(const float* base, unsigned goff, void* lds) {
  asm volatile("global_load_async_to_lds_b128 %0, %1, %2"
               :: "v"((unsigned)(size_t)lds), "v"(goff), "s"(base) : "memory");
}
__device__ __forceinline__ void wait_async0() {
#if __has_builtin(__builtin_amdgcn_s_wait_asynccnt)
  __builtin_amdgcn_s_wait_asynccnt((short)0);
#else
  asm volatile("s_wait_asynccnt 0x0" ::: "memory");
#endif
}

// ---------------------------------------------------------------------------
// Core 32xK @ KxNout GEMM accumulation for one 16x16 output tile per wave.
// 8 waves arranged as 2 row-tiles x 4 col-tiles covering 32 rows x 64 cols
// (one column-group). Weight panels (32 x 64) DMA'd into LDS via async loads.
// Must be called by all 256 threads with uniform arguments.
// ---------------------------------------------------------------------------
__device__ __forceinline__ v8f gemm_acc(const float* sSrc, int srcStride,
                                        const float* __restrict__ W, int Nout, int Kdim,
                                        float* sWp, int nbase,
                                        int t, int m, int khi, int rt, int ct) {
  v8f acc = {};
  for (int kp = 0; kp < Kdim; kp += 32) {
    // async stage panel W[kp:kp+32, nbase:nbase+64] -> sWp (512 float4, 2/thread)
#pragma unroll
    for (int ii = 0; ii < 2; ++ii) {
      int i = t + ii * 256;            // float4 index; kk = i>>4, j4 = i&15
      int kk = i >> 4, j4 = i & 15;
      unsigned goff = (unsigned)(((kp + kk) * Nout + nbase + (j4 << 2)) * 4);
      async_ld_b128_gvs(W, goff, sWp + i * 4);
    }
    if (kp + 32 < Kdim)
      __builtin_prefetch(W + (size_t)(kp + 32) * Nout + nbase + (t & 63), 0, 1);
    wait_async0();
    __syncthreads();
    const float* aRow = sSrc + (rt * 16 + m) * srcStride + kp + khi;
    const float* bCol = sWp + khi * 64 + ct * 16 + m;
#pragma unroll
    for (int k4 = 0; k4 < 8; ++k4) {
      v2f a = { aRow[k4 * 4], aRow[k4 * 4 + 1] };
      v2f b = { bCol[(k4 * 4) * 64], bCol[(k4 * 4 + 1) * 64] };
      acc = wmma_f32(acc, a, b);
    }
    __syncthreads();
  }
  return acc;
}

// ---------------------------------------------------------------------------
// Kernel E: s[j] = sum_{r<256} Wp1[256+r][j]   (rank-1 folding of agg)
// ---------------------------------------------------------------------------
__global__ void colsum_kernel(const float* __restrict__ Wp1, float* __restrict__ sVec) {
  int j = blockIdx.x * blockDim.x + threadIdx.x;
  if (j < 512) {
    float a = 0.f;
    for (int r = 0; r < 256; ++r) a += Wp1[(size_t)(256 + r) * 512 + j];
    sVec[j] = a;
  }
}

// ---------------------------------------------------------------------------
// Kernel C: init per-node scalar aggregate to -inf
// ---------------------------------------------------------------------------
__global__ void init_agg_kernel(unsigned int* __restrict__ agg, int n) {
  int i = blockIdx.x * blockDim.x + threadIdx.x;
  if (i < n) agg[i] = 0xFF800000u;  // -inf
}

// ---------------------------------------------------------------------------
// Stage 32 rows x 256 cols (float4-granular) from global into LDS via async DMA.
// rowBase(e) supplied by caller through per-lane pointer math.
// ---------------------------------------------------------------------------
#define STAGE_256(dstBase, dstOff, srcExpr)                                   \
  for (int ii = 0; ii < 8; ++ii) {                                            \
    int i = threadIdx.x + ii * 256;  /* 2048 float4 */                        \
    int e = i >> 6, c4 = i & 63;                                              \
    async_ld_b128_gv((srcExpr) + (c4 << 2),                                   \
                     (dstBase) + e * SA_STRIDE + (dstOff) + (c4 << 2));       \
  }

// ---------------------------------------------------------------------------
// Kernel A: per-edge  gcn_edge_feature = relu([x_i|ef|x_j]@We1+be1)@We2+be2
//           balance   = sigmoid(...LN/ReLU MLP on ef...)
// ---------------------------------------------------------------------------
__global__ __launch_bounds__(256) void edge_gcn_balance_kernel(
    const float* __restrict__ x, const float* __restrict__ ef,
    const int* __restrict__ eidx,
    const float* __restrict__ We1, const float* __restrict__ be1,
    const float* __restrict__ We2, const float* __restrict__ be2,
    const float* __restrict__ Wb1, const float* __restrict__ bb1,
    const float* __restrict__ g1v, const float* __restrict__ beta1,
    const float* __restrict__ Wb2, const float* __restrict__ bb2,
    const float* __restrict__ g2v, const float* __restrict__ beta2,
    const float* __restrict__ Wb3, const float* __restrict__ bb3,
    float* __restrict__ out_gcn, float* __restrict__ out_bal, int Etot) {
  __shared__ __align__(16) float sA[TE * SA_STRIDE];   // [x_i | ef | x_j]
  __shared__ __align__(16) float sH[TE * SH_STRIDE];   // h1, later balance scratch
  __shared__ __align__(16) float sW[32 * 64];
  __shared__ int sII[TE], sJJ[TE];

  const int e0 = blockIdx.x * TE;
  const int t = threadIdx.x;
  const int lane = t & 31, wave = t >> 5;
  const int m = lane & 15;
  const int khi = (lane >> 4) << 1;  // 0 or 2
  const int hi8 = (lane >> 4) << 3;  // 0 or 8
  const int rt = wave >> 2, ct = wave & 3;

  if (t < TE) { sII[t] = eidx[e0 + t]; sJJ[t] = eidx[Etot + e0 + t]; }
  __syncthreads();

  STAGE_256(sA, 0,   x + (size_t)sII[e] * 256)
  STAGE_256(sA, 256, ef + (size_t)(e0 + e) * 256)
  STAGE_256(sA, 512, x + (size_t)sJJ[e] * 256)
  wait_async0();
  __syncthreads();

  // GEMM1: 32x768 @ 768x512, +bias, relu -> sH
  for (int cg = 0; cg < 8; ++cg) {
    v8f acc = gemm_acc(sA, SA_STRIDE, We1, 512, 768, sW, cg * 64, t, m, khi, rt, ct);
    int col = cg * 64 + ct * 16 + m;
    float bcol = be1[col];
#pragma unroll
    for (int r = 0; r < 8; ++r) {
      float v = acc[r] + bcol;
      sH[(rt * 16 + r + hi8) * SH_STRIDE + col] = v > 0.f ? v : 0.f;
    }
  }
  // GEMM2: 32x512 @ 512x256, +bias -> global
  for (int cg = 0; cg < 4; ++cg) {
    v8f acc = gemm_acc(sH, SH_STRIDE, We2, 256, 512, sW, cg * 64, t, m, khi, rt, ct);
    int col = cg * 64 + ct * 16 + m;
    float bcol = be2[col];
#pragma unroll
    for (int r = 0; r < 8; ++r)
      out_gcn[(size_t)(e0 + rt * 16 + r + hi8) * 256 + col] = acc[r] + bcol;
  }
  __syncthreads();

  // ---- balance MLP (reuse sH as scratch; A = ef slice of sA) ----
  float* sB1 = sH;             // 32 x 132
  float* sB2 = sH + TE * 132;  // 32 x 68
  // b1 = ef @ Wb1 + bb1   (32x256 @ 256x128), B fragments direct from global
  for (int jj = 0; jj < 2; ++jj) {
    int job = wave * 2 + jj;           // 16 jobs = 2 rt x 8 ct
    int brt = job >> 3, bct = job & 7;
    int n = bct * 16 + m;
    const float* aRow = &sA[(brt * 16 + m) * SA_STRIDE + 256 + khi];
    v8f acc = {};
#pragma unroll 4
    for (int k4 = 0; k4 < 64; ++k4) {
      int kk = k4 * 4 + khi;
      v2f a = { aRow[k4 * 4], aRow[k4 * 4 + 1] };
      v2f b = { Wb1[kk * 128 + n], Wb1[(kk + 1) * 128 + n] };
      acc = wmma_f32(acc, a, b);
    }
#pragma unroll
    for (int r = 0; r < 8; ++r)
      sB1[(brt * 16 + r + hi8) * 132 + n] = acc[r] + bb1[n];
  }
  __syncthreads();
  if (t < TE) {  // LayerNorm(128) + relu
    float mu = 0.f, s2 = 0.f;
    for (int j = 0; j < 128; ++j) { float v = sB1[t * 132 + j]; mu += v; s2 += v * v; }
    mu *= (1.f / 128.f); s2 = s2 * (1.f / 128.f) - mu * mu;
    float inv = rsqrtf(s2 + 1e-5f);
    for (int j = 0; j < 128; ++j) {
      float v = (sB1[t * 132 + j] - mu) * inv * g1v[j] + beta1[j];
      sB1[t * 132 + j] = v > 0.f ? v : 0.f;
    }
  }
  __syncthreads();
  {  // b2 = b1 @ Wb2 + bb2  (32x128 @ 128x64); 8 jobs, one per wave
    int brt = wave >> 2, bct = wave & 3;
    int n = bct * 16 + m;
    const float* aRow = &sB1[(brt * 16 + m) * 132 + khi];
    v8f acc = {};
#pragma unroll 4
    for (int k4 = 0; k4 < 32; ++k4) {
      int kk = k4 * 4 + khi;
      v2f a = { aRow[k4 * 4], aRow[k4 * 4 + 1] };
      v2f b = { Wb2[kk * 64 + n], Wb2[(kk + 1) * 64 + n] };
      acc = wmma_f32(acc, a, b);
    }
#pragma unroll
    for (int r = 0; r < 8; ++r)
      sB2[(brt * 16 + r + hi8) * 68 + n] = acc[r] + bb2[n];
  }
  __syncthreads();
  if (t < TE) {  // LN(64) + relu + final dot + sigmoid
    float mu = 0.f, s2 = 0.f;
    for (int j = 0; j < 64; ++j) { float v = sB2[t * 68 + j]; mu += v; s2 += v * v; }
    mu *= (1.f / 64.f); s2 = s2 * (1.f / 64.f) - mu * mu;
    float inv = rsqrtf(s2 + 1e-5f);
    float z = bb3[0];
    for (int j = 0; j < 64; ++j) {
      float v = (sB2[t * 68 + j] - mu) * inv * g2v[j] + beta2[j];
      v = v > 0.f ? v : 0.f;
      z += v * Wb3[j];
    }
    out_bal[e0 + t] = 1.f / (1.f + __expf(-z));
  }
}

// ---------------------------------------------------------------------------
// Kernel B: projections + dual-attention conv nets + softmax + blend + scatter
// ---------------------------------------------------------------------------
__global__ __launch_bounds__(256) void edge_attn_kernel(
    const float* __restrict__ x, const float* __restrict__ ef,
    const float* __restrict__ geo, const int* __restrict__ eidx,
    const float* __restrict__ Wq, const float* __restrict__ bq,
    const float* __restrict__ Wep, const float* __restrict__ bep,
    const float* __restrict__ Wv, const float* __restrict__ bv,
    const float* __restrict__ Wg, const float* __restrict__ bg,
    const float* __restrict__ Wg1, const float* __restrict__ bg1,
    const float* __restrict__ Wg2, const float* __restrict__ bg2,
    const float* __restrict__ Ws1, const float* __restrict__ bs1,
    const float* __restrict__ Ws2, const float* __restrict__ bs2,
    const float* __restrict__ out_bal, float* __restrict__ agg, int Etot) {
  __shared__ __align__(16) float sIn[TE * SA_STRIDE];   // [x_i|ef|x_j]; later conv-L1 acts [e][c][h]
  __shared__ __align__(16) float sQ[TE * SP_STRIDE];
  __shared__ __align__(16) float sEp[TE * SP_STRIDE];
  __shared__ __align__(16) float sV[TE * SP_STRIDE];
  __shared__ __align__(16) float sGp[TE * SP_STRIDE];
  __shared__ __align__(16) float sPr[TE * SP_STRIDE];   // prob buffer (geo, then sem)
  __shared__ __align__(16) float sW2[32 * 64];
  __shared__ float sGeo[TE * 12];
  __shared__ float sOutG[TE], sOutS[TE];
  __shared__ int sII[TE], sJJ[TE];

  const int e0 = blockIdx.x * TE;
  const int t = threadIdx.x;
  const int lane = t & 31, wave = t >> 5;
  const int m = lane & 15;
  const int khi = (lane >> 4) << 1;
  const int hi8 = (lane >> 4) << 3;
  const int rt = wave >> 2, ct = wave & 3;

  if (t < TE) { sII[t] = eidx[e0 + t]; sJJ[t] = eidx[Etot + e0 + t]; }
  __syncthreads();

  STAGE_256(sIn, 0,   x + (size_t)sII[e] * 256)
  STAGE_256(sIn, 256, ef + (size_t)(e0 + e) * 256)
  STAGE_256(sIn, 512, x + (size_t)sJJ[e] * 256)
  for (int i = t; i < TE * 11; i += 256) {
    int e = i / 11, k = i - e * 11;
    sGeo[e * 12 + k] = geo[(size_t)(e0 + e) * 11 + k];
  }
  wait_async0();
  __syncthreads();

  // projections (32x256 @ 256x256 each)
  for (int cg = 0; cg < 4; ++cg) {
    v8f acc = gemm_acc(sIn, SA_STRIDE, Wq, 256, 256, sW2, cg * 64, t, m, khi, rt, ct);
    int col = cg * 64 + ct * 16 + m; float bc = bq[col];
#pragma unroll
    for (int r = 0; r < 8; ++r) sQ[(rt * 16 + r + hi8) * SP_STRIDE + col] = acc[r] + bc;
  }
  for (int cg = 0; cg < 4; ++cg) {
    v8f acc = gemm_acc(sIn + 256, SA_STRIDE, Wep, 256, 256, sW2, cg * 64, t, m, khi, rt, ct);
    int col = cg * 64 + ct * 16 + m; float bc = bep[col];
#pragma unroll
    for (int r = 0; r < 8; ++r) sEp[(rt * 16 + r + hi8) * SP_STRIDE + col] = acc[r] + bc;
  }
  for (int cg = 0; cg < 4; ++cg) {
    v8f acc = gemm_acc(sIn + 512, SA_STRIDE, Wv, 256, 256, sW2, cg * 64, t, m, khi, rt, ct);
    int col = cg * 64 + ct * 16 + m; float bc = bv[col];
#pragma unroll
    for (int r = 0; r < 8; ++r) sV[(rt * 16 + r + hi8) * SP_STRIDE + col] = acc[r] + bc;
  }
  // geo projection (K=11, scalar)
  for (int i = t; i < TE * 256; i += 256) {
    int e = i >> 8, n = i & 255;
    float acc = bg[n];
#pragma unroll
    for (int k = 0; k < 11; ++k) acc += sGeo[e * 12 + k] * Wg[k * 256 + n];
    sGp[e * SP_STRIDE + n] = acc;
  }
  __syncthreads();

  // geo conv L1: per-h GEMM (32x96 @ 96x96), relu -> sIn as [e][c][h]
  for (int it = 0; it < 12; ++it) {
    int h = wave, brt = it / 6, bct = it - brt * 6;
    int e_ = brt * 16 + m, o_ = bct * 16 + m;
    v8f acc = {};
#pragma unroll 4
    for (int k4 = 0; k4 < 24; ++k4) {
      int kk = k4 * 4 + khi;
      float ax, ay;
      if (kk < 32)      { ax = sQ[e_ * SP_STRIDE + kk * 8 + h];        ay = sQ[e_ * SP_STRIDE + (kk + 1) * 8 + h]; }
      else if (kk < 64) { ax = sEp[e_ * SP_STRIDE + (kk - 32) * 8 + h]; ay = sEp[e_ * SP_STRIDE + (kk - 31) * 8 + h]; }
      else              { ax = sGp[e_ * SP_STRIDE + (kk - 64) * 8 + h]; ay = sGp[e_ * SP_STRIDE + (kk - 63) * 8 + h]; }
      v2f a = { ax, ay };
      v2f b = { Wg1[o_ * 96 + kk], Wg1[o_ * 96 + kk + 1] };
      acc = wmma_f32(acc, a, b);
    }
#pragma unroll
    for (int r = 0; r < 8; ++r) {
      float v = acc[r] + bg1[o_];
      sIn[(brt * 16 + r + hi8) * SA_STRIDE + o_ * 8 + h] = v > 0.f ? v : 0.f;
    }
  }
  __syncthreads();
  // geo conv L2: (32x96 @ 96x32) -> sPr
  for (int it = 0; it < 4; ++it) {
    int h = wave, brt = it >> 1, bct = it & 1;
    int e_ = brt * 16 + m, o_ = bct * 16 + m;
    v8f acc = {};
#pragma unroll 4
    for (int k4 = 0; k4 < 24; ++k4) {
      int kk = k4 * 4 + khi;
      v2f a = { sIn[e_ * SA_STRIDE + kk * 8 + h], sIn[e_ * SA_STRIDE + (kk + 1) * 8 + h] };
      v2f b = { Wg2[o_ * 96 + kk], Wg2[o_ * 96 + kk + 1] };
      acc = wmma_f32(acc, a, b);
    }
#pragma unroll
    for (int r = 0; r < 8; ++r)
      sPr[(brt * 16 + r + hi8) * SP_STRIDE + o_ * 8 + h] = acc[r] + bg2[o_];
  }
  __syncthreads();
  {  // softmax over 32 channels per (e,h)
    int e = t >> 3, h = t & 7;
    float mx = -1e30f;
    for (int o = 0; o < 32; ++o) mx = fmaxf(mx, sPr[e * SP_STRIDE + o * 8 + h]);
    float sum = 0.f;
    for (int o = 0; o < 32; ++o) { float p = __expf(sPr[e * SP_STRIDE + o * 8 + h] - mx); sPr[e * SP_STRIDE + o * 8 + h] = p; sum += p; }
    float inv = 1.f / sum;
    for (int o = 0; o < 32; ++o) sPr[e * SP_STRIDE + o * 8 + h] *= inv;
  }
  __syncthreads();
  if (t < TE) {  // geo_out[e] = sum prob*value  ( [o][h] flattens to plain 256 cols )
    float s = 0.f;
    for (int i = 0; i < 256; ++i) s += sPr[t * SP_STRIDE + i] * sV[t * SP_STRIDE + i];
    sOutG[t] = s;
  }
  __syncthreads();

  // sem conv L1: (32x64 @ 64x64), relu -> sIn
  for (int it = 0; it < 8; ++it) {
    int h = wave, brt = it >> 2, bct = it & 3;
    int e_ = brt * 16 + m, o_ = bct * 16 + m;
    v8f acc = {};
#pragma unroll 4
    for (int k4 = 0; k4 < 16; ++k4) {
      int kk = k4 * 4 + khi;
      float ax, ay;
      if (kk < 32) { ax = sQ[e_ * SP_STRIDE + kk * 8 + h];         ay = sQ[e_ * SP_STRIDE + (kk + 1) * 8 + h]; }
      else         { ax = sEp[e_ * SP_STRIDE + (kk - 32) * 8 + h]; ay = sEp[e_ * SP_STRIDE + (kk - 31) * 8 + h]; }
      v2f a = { ax, ay };
      v2f b = { Ws1[o_ * 64 + kk], Ws1[o_ * 64 + kk + 1] };
      acc = wmma_f32(acc, a, b);
    }
#pragma unroll
    for (int r = 0; r < 8; ++r) {
      float v = acc[r] + bs1[o_];
      sIn[(brt * 16 + r + hi8) * SA_STRIDE + o_ * 8 + h] = v > 0.f ? v : 0.f;
    }
  }
  __syncthreads();
  // sem conv L2: (32x64 @ 64x32) -> sPr
  for (int it = 0; it < 4; ++it) {
    int h = wave, brt = it >> 1, bct = it & 1;
    int e_ = brt * 16 + m, o_ = bct * 16 + m;
    v8f acc = {};
#pragma unroll 4
    for (int k4 = 0; k4 < 16; ++k4) {
      int kk = k4 * 4 + khi;
      v2f a = { sIn[e_ * SA_STRIDE + kk * 8 + h], sIn[e_ * SA_STRIDE + (kk + 1) * 8 + h] };
      v2f b = { Ws2[o_ * 64 + kk], Ws2[o_ * 64 + kk + 1] };
      acc = wmma_f32(acc, a, b);
    }
#pragma unroll
    for (int r = 0; r < 8; ++r)
      sPr[(brt * 16 + r + hi8) * SP_STRIDE + o_ * 8 + h] = acc[r] + bs2[o_];
  }
  __syncthreads();
  {  // softmax (sem)
    int e = t >> 3, h = t & 7;
    float mx = -1e30f;
    for (int o = 0; o < 32; ++o) mx = fmaxf(mx, sPr[e * SP_STRIDE + o * 8 + h]);
    float sum = 0.f;
    for (int o = 0; o < 32; ++o) { float p = __expf(sPr[e * SP_STRIDE + o * 8 + h] - mx); sPr[e * SP_STRIDE + o * 8 + h] = p; sum += p; }
    float inv = 1.f / sum;
    for (int o = 0; o < 32; ++o) sPr[e * SP_STRIDE + o * 8 + h] *= inv;
  }
  __syncthreads();
  if (t < TE) {
    float s = 0.f;
    for (int i = 0; i < 256; ++i) s += sPr[t * SP_STRIDE + i] * sV[t * SP_STRIDE + i];
    sOutS[t] = s;
    float bal = out_bal[e0 + t];
    float sc = bal * sOutG[t] + (1.f - bal) * sOutS[t];
    atomicMaxFloat(&agg[sII[t]], sc);
  }
}

// ---------------------------------------------------------------------------
// Kernel D: node prop. h=[x|agg*1]; t=relu(x@Wp1_top + agg*s + bp1); out=t@Wp2+bp2
// ---------------------------------------------------------------------------
__global__ __launch_bounds__(256) void node_prop_kernel(
    const float* __restrict__ x, const float* __restrict__ Wp1,
    const float* __restrict__ bp1, const float* __restrict__ Wp2,
    const float* __restrict__ bp2, const float* __restrict__ sVec,
    const float* __restrict__ agg, float* __restrict__ out_xx) {
  __shared__ __align__(16) float sX[TE * SP_STRIDE];
  __shared__ __align__(16) float sT[TE * SH_STRIDE];
  __shared__ __align__(16) float sW[32 * 64];
  __shared__ float sS[512];
  __shared__ float sAgg[TE];

  const int n0 = blockIdx.x * TE;
  const int t = threadIdx.x;
  const int lane = t & 31, wave = t >> 5;
  const int m = lane & 15;
  const int khi = (lane >> 4) << 1;
  const int hi8 = (lane >> 4) << 3;
  const int rt = wave >> 2, ct = wave & 3;

  if (t < TE) { float a = agg[n0 + t]; sAgg[t] = (a < -3.0e38f) ? 0.f : a; }
  for (int i = t; i < 512; i += 256) sS[i] = sVec[i];
  // async stage x tile (32x256 = 2048 float4)
  {
    const float* xb = x + (size_t)n0 * 256;
#pragma unroll
    for (int ii = 0; ii < 8; ++ii) {
      int i = t + ii * 256;
      int e = i >> 6, c4 = i & 63;
      unsigned goff = (unsigned)((e * 256 + (c4 << 2)) * 4);
      async_ld_b128_gvs(xb, goff, sX + e * SP_STRIDE + (c4 << 2));
    }
  }
  wait_async0();
  __syncthreads();

  for (int cg = 0; cg < 8; ++cg) {  // 32x256 @ 256x512 (top half of Wp1)
    v8f acc = gemm_acc(sX, SP_STRIDE, Wp1, 512, 256, sW, cg * 64, t, m, khi, rt, ct);
    int col = cg * 64 + ct * 16 + m;
    float bc = bp1[col], sc = sS[col];
#pragma unroll
    for (int r = 0; r < 8; ++r) {
      int row = rt * 16 + r + hi8;
      float v = acc[r] + sAgg[row] * sc + bc;
      sT[row * SH_STRIDE + col] = v > 0.f ? v : 0.f;
    }
  }
  for (int cg = 0; cg < 4; ++cg) {  // 32x512 @ 512x256
    v8f acc = gemm_acc(sT, SH_STRIDE, Wp2, 256, 512, sW, cg * 64, t, m, khi, rt, ct);
    int col = cg * 64 + ct * 16 + m;
    float bc = bp2[col];
#pragma unroll
    for (int r = 0; r < 8; ++r)
      out_xx[(size_t)(n0 + rt * 16 + r + hi8) * 256 + col] = acc[r] + bc;
  }
}

// ---------------------------------------------------------------------------
extern "C" void kernel_launch(void* const* d_in, const int* in_sizes, int n_in,
                              void* d_out, int out_size, void* d_ws, size_t ws_size,
                              hipStream_t stream) {
  const float* x    = (const float*)d_in[0];
  const float* ef   = (const float*)d_in[1];
  const float* geo  = (const float*)d_in[2];
  const int*   eidx = (const int*)d_in[3];
  const float* We1 = (const float*)d_in[4],  *be1 = (const float*)d_in[5];
  const float* We2 = (const float*)d_in[6],  *be2 = (const float*)d_in[7];
  const float* Wq  = (const float*)d_in[8],  *bq  = (const float*)d_in[9];
  const float* Wep = (const float*)d_in[10], *bep = (const float*)d_in[11];
  const float* Wv  = (const float*)d_in[12], *bv  = (const float*)d_in[13];
  const float* Wg  = (const float*)d_in[14], *bg  = (const float*)d_in[15];
  const float* Wg1 = (const float*)d_in[16], *bg1 = (const float*)d_in[17];
  const float* Wg2 = (const float*)d_in[18], *bg2 = (const float*)d_in[19];
  const float* Ws1 = (const float*)d_in[20], *bs1 = (const float*)d_in[21];
  const float* Ws2 = (const float*)d_in[22], *bs2 = (const float*)d_in[23];
  const float* Wb1 = (const float*)d_in[24], *bb1 = (const float*)d_in[25];
  const float* g1  = (const float*)d_in[26], *bt1 = (const float*)d_in[27];
  const float* Wb2 = (const float*)d_in[28], *bb2 = (const float*)d_in[29];
  const float* g2  = (const float*)d_in[30], *bt2 = (const float*)d_in[31];
  const float* Wb3 = (const float*)d_in[32], *bb3 = (const float*)d_in[33];
  const float* Wp1 = (const float*)d_in[34], *bp1 = (const float*)d_in[35];
  const float* Wp2 = (const float*)d_in[36], *bp2 = (const float*)d_in[37];

  const int N = in_sizes[0] / 256;
  const int E = in_sizes[3] / 2;

  float* out_xx  = (float*)d_out;                        // (N, 256)
  float* out_gcn = out_xx + (size_t)N * 256;             // (E, 256)
  float* out_bal = out_gcn + (size_t)E * 256;            // (E,)

  float* sVec = (float*)d_ws;                            // 512 floats
  float* aggs = sVec + 512;                              // N floats

  colsum_kernel<<<2, 256, 0, stream>>>(Wp1, sVec);
  init_agg_kernel<<<(N + 255) / 256, 256, 0, stream>>>((unsigned int*)aggs, N);

  edge_gcn_balance_kernel<<<E / TE, 256, 0, stream>>>(
      x, ef, eidx, We1, be1, We2, be2,
      Wb1, bb1, g1, bt1, Wb2, bb2, g2, bt2, Wb3, bb3,
      out_gcn, out_bal, E);

  edge_attn_kernel<<<E / TE, 256, 0, stream>>>(
      x, ef, geo, eidx, Wq, bq, Wep, bep, Wv, bv, Wg, bg,
      Wg1, bg1, Wg2, bg2, Ws1, bs1, Ws2, bs2,
      out_bal, aggs, E);

  node_prop_kernel<<<N / TE, 256, 0, stream>>>(
      x, Wp1, bp1, Wp2, bp2, sVec, aggs, out_xx);
}